// LocalAttention_73065983639704
// MI455X (gfx1250) — compile-verified
//
#include <hip/hip_runtime.h>
#include <hip/hip_bf16.h>

// Local (sliding-window) attention, fp32, via V_WMMA_F32_16X16X4_F32.
// B=2, S=2048, E=256, pad=128, W=257.
//
// Workspace layout (floats):
//   q     : [4096  x 256]   (B*S rows)
//   k_ext : [4608  x 256]   (B*(S+2p) rows, pad rows == bias)
//   v_ext : [4608  x 256]
// total = 13.63 MB

typedef __attribute__((ext_vector_type(2))) float v2f;
typedef __attribute__((ext_vector_type(8))) float v8f;

#define EDIM   256
#define SLEN   2048
#define NBATCH 2
#define PADR   128
#define KEXT_ROWS 2304              // SLEN + 2*PADR
#define Q_ROWS    4096              // NBATCH*SLEN
#define KV_ROWS   4608              // NBATCH*KEXT_ROWS
#define LDS_STRIDE 276              // 272 cols + 4 pad -> conflict-free transpose reads

static __device__ __forceinline__ v8f wmma_f32(v2f a, v2f b, v8f c) {
    return __builtin_amdgcn_wmma_f32_16x16x4_f32(false, a, false, b, (short)0, c,
                                                 false, false);
}

// ---------------------------------------------------------------------------
// Kernel 1: fused Q/K/V projections.  One wave = 16 rows x 64 cols of one of
// {q, k_ext, v_ext}.  K-loop: 64 x V_WMMA_F32_16X16X4_F32 per 16x16 tile.
//   jobs: q   : 256 row-tiles * 4 colgroups = 1024
//         k   : 288 row-tiles * 4 colgroups = 1152   (padded row space)
//         v   : 1152
//   total 3328 wave-jobs; 8 waves/block -> 416 blocks of 256 threads.
//
// Pad rows (zero rows of x_ext) are handled branch-free: the A-fragment load
// reads a clamped valid address unconditionally and is multiplied by a
// loop-invariant {0,1} scale, so the inner loop has no EXEC manipulation.
// ---------------------------------------------------------------------------
__global__ void qkv_proj_kernel(const float* __restrict__ x,
                                const float* __restrict__ Wq, const float* __restrict__ bq,
                                const float* __restrict__ Wk, const float* __restrict__ bk,
                                const float* __restrict__ Wv, const float* __restrict__ bv,
                                float* __restrict__ q,
                                float* __restrict__ kext,
                                float* __restrict__ vext)
{
    const int lane = threadIdx.x & 31;
    const int job  = blockIdx.x * (blockDim.x >> 5) + (threadIdx.x >> 5);

    const float* W;  const float* bias;  float* outp;
    int rt, cg;      bool padded;
    if (job < 1024)      { int t = job;        W = Wq; bias = bq; outp = q;    rt = t >> 2; cg = t & 3; padded = false; }
    else if (job < 2176) { int t = job - 1024; W = Wk; bias = bk; outp = kext; rt = t >> 2; cg = t & 3; padded = true;  }
    else                 { int t = job - 2176; W = Wv; bias = bv; outp = vext; rt = t >> 2; cg = t & 3; padded = true;  }

    const int nlo = lane & 15;          // A-row / B-col within tile
    const int hf  = lane >> 4;          // half-wave selects K pair
    const int kb  = hf * 2;

    // A-fragment source row for this lane (constant across the K loop)
    const int R = rt * 16 + nlo;
    int  xrow;
    float scale;
    if (!padded) {
        xrow  = R;                                  // q rows map 1:1 to x rows
        scale = 1.0f;
    } else {
        const int bb = R / KEXT_ROWS;
        const int sr = R - bb * KEXT_ROWS;
        const int xr = sr - PADR;
        const bool valid = (xr >= 0) && (xr < SLEN);
        // clamp to a valid row; contribution zeroed by scale
        const int xc = valid ? xr : 0;
        xrow  = bb * SLEN + xc;
        scale = valid ? 1.0f : 0.0f;
    }
    const float* xp = x + (size_t)xrow * EDIM;

    v8f acc[4] = {};
    for (int kc = 0; kc < 64; ++kc) {
        const int k0 = kc * 4;
        v2f a;
        a.x = xp[k0 + kb]     * scale;
        a.y = xp[k0 + kb + 1] * scale;
#pragma unroll
        for (int nt = 0; nt < 4; ++nt) {
            const int N = cg * 64 + nt * 16 + nlo;
            v2f b;
            b.x = W[(size_t)(k0 + kb)     * EDIM + N];
            b.y = W[(size_t)(k0 + kb + 1) * EDIM + N];
            acc[nt] = wmma_f32(a, b, acc[nt]);
        }
    }

    // C/D layout: lane holds rows (r + 8*hf), col = nt*16 + nlo
#pragma unroll
    for (int nt = 0; nt < 4; ++nt) {
        const int col = cg * 64 + nt * 16 + nlo;
        const float bval = bias[col];
#pragma unroll
        for (int r = 0; r < 8; ++r) {
            const int Row = rt * 16 + r + 8 * hf;
            outp[(size_t)Row * EDIM + col] = acc[nt][r] + bval;
        }
    }
}

// ---------------------------------------------------------------------------
// Kernel 2: banded attention.  One wave = 16 query rows of one batch.
// Keys/values needed: k_ext rows [s0, s0+271]  (17 tiles of 16).
// Phase 1: energy = Q K^T / 16 via WMMA, band-masked, staged in LDS,
//          running row-max in registers (reduced with shfl_xor over 16 lanes).
// Phase 2: exp/sum pass through LDS (unnormalized P written back).
// Phase 3: out = P x V via WMMA (A-fragments read transposed from LDS),
//          normalized by row-sum at store.
// 2 waves / block -> 35.3 KB LDS static.
// ---------------------------------------------------------------------------
__global__ void local_attn_kernel(const float* __restrict__ q,
                                  const float* __restrict__ kext,
                                  const float* __restrict__ vext,
                                  float* __restrict__ out)
{
    __shared__ float lds[2 * 16 * LDS_STRIDE];

    const int lane = threadIdx.x & 31;
    const int wid  = threadIdx.x >> 5;
    const int tile = blockIdx.x * 2 + wid;          // 0..255
    const int bb   = tile >> 7;                     // batch
    const int s0   = (tile & 127) * 16;             // first query row

    float* myLds = lds + wid * 16 * LDS_STRIDE;

    const int nlo = lane & 15;
    const int hf  = lane >> 4;
    const int kb  = hf * 2;

    // --- Phase 1: energy = Q K^T * (1/sqrt(E)), band mask, row max ---------
    const float* qrow = q + (size_t)(bb * SLEN + s0 + nlo) * EDIM;   // A row
    float rowmax[8];
#pragma unroll
    for (int r = 0; r < 8; ++r) rowmax[r] = -3.0e38f;

    for (int jt = 0; jt < 17; ++jt) {
        // B column N = nlo -> key row (s0 + jt*16 + nlo); K^T fragment is a
        // contiguous 8-byte read from that key row.
        const float* krow = kext + (size_t)(bb * KEXT_ROWS + s0 + jt * 16 + nlo) * EDIM;
        v8f acc = {};
        for (int kc = 0; kc < 64; ++kc) {
            const int k0 = kc * 4;
            v2f a; a.x = qrow[k0 + kb]; a.y = qrow[k0 + kb + 1];
            v2f b; b.x = krow[k0 + kb]; b.y = krow[k0 + kb + 1];
            acc = wmma_f32(a, b, acc);
        }
        const int j = jt * 16 + nlo;                // key index 0..271
#pragma unroll
        for (int r = 0; r < 8; ++r) {
            const int m = r + 8 * hf;               // query row 0..15
            float e = acc[r] * 0.0625f;             // 1/sqrt(256)
            const bool ok = (j >= m) && (j <= m + 256);   // w in [0,257)
            e = ok ? e : -1.0e30f;
            rowmax[r] = fmaxf(rowmax[r], e);
            myLds[m * LDS_STRIDE + j] = e;
        }
    }
    // rows 0..7 live in lanes 0..15, rows 8..15 in lanes 16..31:
    // xor masks 1,2,4,8 reduce exactly within each half.
#pragma unroll
    for (int mask = 1; mask < 16; mask <<= 1)
#pragma unroll
        for (int r = 0; r < 8; ++r)
            rowmax[r] = fmaxf(rowmax[r], __shfl_xor(rowmax[r], mask, 32));

    __syncthreads();

    // --- Phase 2: exp + row sum, write unnormalized P back to LDS ----------
    float rowsum[8];
#pragma unroll
    for (int r = 0; r < 8; ++r) rowsum[r] = 0.0f;

    for (int jt = 0; jt < 17; ++jt) {
        const int j = jt * 16 + nlo;
#pragma unroll
        for (int r = 0; r < 8; ++r) {
            const int m = r + 8 * hf;
            const float e = myLds[m * LDS_STRIDE + j];
            const float p = __expf(e - rowmax[r]);
            rowsum[r] += p;
            myLds[m * LDS_STRIDE + j] = p;
        }
    }
#pragma unroll
    for (int mask = 1; mask < 16; mask <<= 1)
#pragma unroll
        for (int r = 0; r < 8; ++r)
            rowsum[r] += __shfl_xor(rowsum[r], mask, 32);

    __syncthreads();

    // --- Phase 3: out = P x V (K = 272 = 68 chunks of 4) --------------------
    v8f oacc[16] = {};
    for (int kc = 0; kc < 68; ++kc) {
        const int k0 = kc * 4;
        // A-fragment of P from LDS (transposed read, stride 276 floats)
        v2f a;
        a.x = myLds[nlo * LDS_STRIDE + k0 + kb];
        a.y = myLds[nlo * LDS_STRIDE + k0 + kb + 1];
        const float* vr = vext + (size_t)(bb * KEXT_ROWS + s0 + k0 + kb) * EDIM;
#pragma unroll
        for (int nt = 0; nt < 16; ++nt) {
            const int col = nt * 16 + nlo;
            v2f b;
            b.x = vr[col];          // V[s0+k0+kb]  [col]
            b.y = vr[EDIM + col];   // V[s0+k0+kb+1][col]
            oacc[nt] = wmma_f32(a, b, oacc[nt]);
        }
    }

    // store normalized output; C-layout rows match rowsum's lane-half rows
#pragma unroll
    for (int nt = 0; nt < 16; ++nt) {
        const int col = nt * 16 + nlo;
#pragma unroll
        for (int r = 0; r < 8; ++r) {
            const int Row = s0 + r + 8 * hf;
            out[(size_t)(bb * SLEN + Row) * EDIM + col] = oacc[nt][r] / rowsum[r];
        }
    }
}

// ---------------------------------------------------------------------------
extern "C" void kernel_launch(void* const* d_in, const int* in_sizes, int n_in,
                              void* d_out, int out_size, void* d_ws, size_t ws_size,
                              hipStream_t stream)
{
    (void)in_sizes; (void)n_in; (void)out_size; (void)ws_size;

    const float* x  = (const float*)d_in[0];
    const float* Wq = (const float*)d_in[1];
    const float* bq = (const float*)d_in[2];
    const float* Wk = (const float*)d_in[3];
    const float* bk = (const float*)d_in[4];
    const float* Wv = (const float*)d_in[5];
    const float* bv = (const float*)d_in[6];
    float* out = (float*)d_out;

    float* ws   = (float*)d_ws;
    float* qbuf = ws;                                   // 4096*256
    float* kbuf = qbuf + (size_t)Q_ROWS  * EDIM;        // 4608*256
    float* vbuf = kbuf + (size_t)KV_ROWS * EDIM;        // 4608*256

    // 3328 wave-jobs, 8 waves (256 threads) per block
    qkv_proj_kernel<<<416, 256, 0, stream>>>(x, Wq, bq, Wk, bk, Wv, bv,
                                             qbuf, kbuf, vbuf);
    // 256 query tiles, 2 waves (64 threads) per block
    local_attn_kernel<<<128, 64, 0, stream>>>(qbuf, kbuf, vbuf, out);
}